// ActionNet_79379585564965
// MI455X (gfx1250) — compile-verified
//
#include <hip/hip_runtime.h>
#include <hip/hip_bf16.h>

typedef float v2f __attribute__((ext_vector_type(2)));
typedef float v8f __attribute__((ext_vector_type(8)));

#define NNODES  5000
#define NEDGES  60000
#define ETOT    65000     // NEDGES + NNODES self loops
#define NHEADS  4
#define NLAYERS 7
#define KC      32        // K-chunk staged in LDS (multiple of 4)

// ---------------------------------------------------------------------------
// Append self loops: src/dst arrays of length ETOT
// ---------------------------------------------------------------------------
__global__ void k_build_edges(const int* __restrict__ ei,
                              int* __restrict__ src, int* __restrict__ dst) {
  int i = blockIdx.x * blockDim.x + threadIdx.x;
  if (i >= ETOT) return;
  if (i < NEDGES) { src[i] = ei[i]; dst[i] = ei[NEDGES + i]; }
  else            { src[i] = i - NEDGES; dst[i] = i - NEDGES; }
}

// ---------------------------------------------------------------------------
// fp32 WMMA GEMM:  Y = X[M,K] @ W[K,Ncols]
//  - block = 8 waves; block tile = 128 rows x 32 cols; wave tile = 16 x 32
//    (two 16x16 WMMA tiles sharing one A fragment -> A-reuse, 2x wmma/load)
//  - B panel (KC x 32, 4 KB) staged in LDS, PRE-SWIZZLED so each lane's
//    {B[k][c], B[k+1][c]} fragment is a contiguous 8B LDS word ->
//    ds_load_b64 straight into the even-aligned VGPR pair (no fix-up movs)
//  - grid = (ceil(N/32 cols), ceil(tiles_m/8), 2); z selects (W0,Y0)/(W1,Y1)
//  - K is a multiple of 4 for every layer (4,128,512,1024,512,256,128)
//  - M/N tails handled by CLAMPING indices (dup row/col, store-guarded):
//    no predicated loads, no divergent barriers, EXEC all-1s at every wmma.
// ---------------------------------------------------------------------------
__global__ __launch_bounds__(256) void k_gemm_wmma(
    const float* __restrict__ X, const float* __restrict__ W0,
    const float* __restrict__ W1, float* __restrict__ Y0,
    float* __restrict__ Y1, int M, int K, int Ncols)
{
  const float* __restrict__ W = blockIdx.z ? W1 : W0;
  float* __restrict__ Y       = blockIdx.z ? Y1 : Y0;

  // swizzled layout: lb[(k/2)*64 + col*2 + (k&1)],  (KC/2)*64 = KC*32 floats
  __shared__ float lb[KC * 32];

  const int tid  = threadIdx.x;
  const int lane = tid & 31;
  const int wave = tid >> 5;
  const int lrow = lane & 15;     // row of A / col of B this lane owns
  const int half = lane >> 4;     // lane halves carry K pairs {0,1} / {2,3}

  const int tiles_m = (M + 15) >> 4;
  const int tn  = blockIdx.x;                      // 32-col supertile
  const int tm  = blockIdx.y * 8 + wave;           // M-tile per wave
  const int tmc = tm < tiles_m ? tm : tiles_m - 1; // clamp (store-guarded)

  const int colbase = tn * 32;
  const int c0 = colbase + lrow;                   // col of first 16x16 tile
  const int c1 = colbase + 16 + lrow;              // col of second 16x16 tile

  const int arow  = tmc * 16 + lrow;
  const int arowc = arow < M ? arow : M - 1;       // clamp (store-guarded)
  const float* __restrict__ Xrow = X + (long)arowc * K;

  v8f acc0 = {};
  v8f acc1 = {};
  for (int kc = 0; kc < K; kc += KC) {
    const int klen = (K - kc) < KC ? (K - kc) : KC;    // multiple of 4

    if (kc + KC < K)                                   // warm next A chunk
      __builtin_prefetch(Xrow + kc + KC, 0, 0);        // -> global_prefetch_b8

    // Cooperative swizzled stage of B[kc .. kc+klen) x 32 cols into LDS
    for (int idx = tid; idx < klen * 32; idx += 256) {
      const int r = idx >> 5;                          // k within chunk
      const int c = idx & 31;                          // col within panel
      const int cg = colbase + c;
      const int cc = cg < Ncols ? cg : Ncols - 1;      // clamp (store-guarded)
      lb[(r >> 1) * 64 + c * 2 + (r & 1)] = W[(long)(kc + r) * Ncols + cc];
    }
    __syncthreads();

#pragma unroll 4
    for (int kk = 0; kk < klen; kk += 4) {
      // A (16x4 f32): lanes 0-15 hold K=kk+{0,1}, lanes 16-31 K=kk+{2,3};
      // both elements contiguous & 8B-aligned -> one global_load_b64
      const v2f a = *(const v2f*)(Xrow + kc + kk + half * 2);
      // B fragments: k-pair p = (kk + half*2)/2; contiguous 8B in LDS
      const int p = (kk >> 1) + half;
      const v2f b0 = *(const v2f*)&lb[p * 64 + lrow * 2];
      const v2f b1 = *(const v2f*)&lb[p * 64 + lrow * 2 + 32];
      acc0 = __builtin_amdgcn_wmma_f32_16x16x4_f32(
          false, a, false, b0, (short)0, acc0, false, false);
      acc1 = __builtin_amdgcn_wmma_f32_16x16x4_f32(
          false, a, false, b1, (short)0, acc1, false, false);
    }
    __syncthreads();
  }

  // C/D layout: VGPR r -> M = 16*tm + r + 8*half, N = c0 / c1
  if (tm < tiles_m) {
    const int rbase = tm * 16 + half * 8;
#pragma unroll
    for (int r = 0; r < 8; ++r) {
      const int rr = rbase + r;
      if (rr < M) {
        if (c0 < Ncols) Y[(long)rr * Ncols + c0] = acc0[r];
        if (c1 < Ncols) Y[(long)rr * Ncols + c1] = acc1[r];
      }
    }
  }
}

// ---------------------------------------------------------------------------
// Zero per-layer accumulators: accum[N*dout], denom[N*H], mkey[N*H]
// ---------------------------------------------------------------------------
__global__ void k_layer_init(float* __restrict__ accum, long na,
                             float* __restrict__ denom,
                             unsigned* __restrict__ mkey, int nh)
{
  const long i = (long)blockIdx.x * blockDim.x + threadIdx.x;
  if (i < na) accum[i] = 0.f;
  if (i < nh) { denom[i] = 0.f; mkey[i] = 0u; }   // key 0 == -inf ordering
}

__device__ __forceinline__ unsigned ord_key(float f) {
  unsigned u = __float_as_uint(f);
  return (u & 0x80000000u) ? ~u : (u | 0x80000000u);
}
__device__ __forceinline__ float ord_dec(unsigned k) {
  unsigned u = (k & 0x80000000u) ? (k & 0x7fffffffu) : ~k;
  return __uint_as_float(u);
}

// ---------------------------------------------------------------------------
// logits[e,h] = sum_c att[h,c] * leaky_relu(xl[src,h,c] + xr[dst,h,c])
// + running segment max (ordered-uint atomicMax). One wave32 per (edge,head).
// ---------------------------------------------------------------------------
__global__ __launch_bounds__(256) void k_logits(
    const float* __restrict__ xl, const float* __restrict__ xr,
    const float* __restrict__ att, const int* __restrict__ src,
    const int* __restrict__ dst, float* __restrict__ logits,
    unsigned* __restrict__ mkey, int C)
{
  const int lane = threadIdx.x & 31;
  const long wid = ((long)blockIdx.x * blockDim.x + threadIdx.x) >> 5;
  if (wid >= (long)ETOT * NHEADS) return;               // wave-uniform
  const int e = (int)(wid >> 2);
  const int h = (int)(wid & 3);
  const int s = src[e], d = dst[e];
  const float* __restrict__ pl = xl + ((long)s * NHEADS + h) * C;
  const float* __restrict__ pr = xr + ((long)d * NHEADS + h) * C;
  const float* __restrict__ pav = att + (long)h * C;
  float part = 0.f;
  for (int c = lane; c < C; c += 32) {
    float v = pl[c] + pr[c];
    v = v > 0.f ? v : 0.2f * v;                         // negative_slope 0.2
    part += pav[c] * v;
  }
  for (int o = 16; o > 0; o >>= 1) part += __shfl_down(part, o, 32);
  if (lane == 0) {
    logits[wid] = part;
    atomicMax(&mkey[(long)d * NHEADS + h], ord_key(part));
  }
}

// ---------------------------------------------------------------------------
// ex[e,h] = exp(logit - max[dst,h]);  denom[dst,h] += ex
// ---------------------------------------------------------------------------
__global__ void k_exp(const float* __restrict__ logits,
                      const unsigned* __restrict__ mkey,
                      float* __restrict__ denom, const int* __restrict__ dst,
                      float* __restrict__ exv)
{
  const long i = (long)blockIdx.x * blockDim.x + threadIdx.x;
  if (i >= (long)ETOT * NHEADS) return;
  const int e = (int)(i >> 2);
  const int h = (int)(i & 3);
  const int d = dst[e];
  const float m = ord_dec(mkey[(long)d * NHEADS + h]);
  const float ex = expf(logits[i] - m);
  exv[i] = ex;
  atomicAdd(&denom[(long)d * NHEADS + h], ex);
}

// ---------------------------------------------------------------------------
// accum[dst,c] += alpha * xl[src,h,c]  (head-mean folded in at finalize;
// mean over heads is linear so we accumulate all heads into [N,dout])
// ---------------------------------------------------------------------------
__global__ __launch_bounds__(256) void k_aggr(
    const float* __restrict__ xl, const float* __restrict__ exv,
    const float* __restrict__ denom, const int* __restrict__ src,
    const int* __restrict__ dst, float* __restrict__ accum, int C)
{
  const int lane = threadIdx.x & 31;
  const long wid = ((long)blockIdx.x * blockDim.x + threadIdx.x) >> 5;
  if (wid >= (long)ETOT * NHEADS) return;
  const int e = (int)(wid >> 2);
  const int h = (int)(wid & 3);
  const int s = src[e], d = dst[e];
  const float alpha = exv[wid] / (denom[(long)d * NHEADS + h] + 1e-16f);
  const float* __restrict__ pl = xl + ((long)s * NHEADS + h) * C;
  float* __restrict__ pacc = accum + (long)d * C;
  for (int c = lane; c < C; c += 32)
    atomicAdd(&pacc[c], pl[c] * alpha);
}

// ---------------------------------------------------------------------------
// out[n,c] = act(accum[n,c]/H + b[c])   (ReLU, sigmoid on last layer)
// ---------------------------------------------------------------------------
__global__ void k_final(const float* __restrict__ accum,
                        const float* __restrict__ b, float* __restrict__ out,
                        long total, int C, int last)
{
  const long i = (long)blockIdx.x * blockDim.x + threadIdx.x;
  if (i >= total) return;
  const int c = (int)(i % C);
  float v = accum[i] * (1.f / NHEADS) + b[c];
  out[i] = last ? (1.f / (1.f + expf(-v))) : (v > 0.f ? v : 0.f);
}

// ---------------------------------------------------------------------------
// Host driver
// ---------------------------------------------------------------------------
static const int DIN[NLAYERS]  = {4, 128, 512, 1024, 512, 256, 128};
static const int DOUT[NLAYERS] = {128, 512, 1024, 512, 256, 128, 1};

extern "C" void kernel_launch(void* const* d_in, const int* in_sizes, int n_in,
                              void* d_out, int out_size, void* d_ws,
                              size_t ws_size, hipStream_t stream)
{
  (void)n_in; (void)out_size; (void)ws_size;

  const float* x;
  const int* ei;
  if (in_sizes[0] == 2 * NEDGES) {            // defensive: x/edge order swap
    ei = (const int*)d_in[0]; x = (const float*)d_in[1];
  } else {
    x = (const float*)d_in[0]; ei = (const int*)d_in[1];
  }

  // -------- workspace carve (256B aligned) ---------------------------------
  char* wsp = (char*)d_ws;
  auto carve = [&](size_t bytes) -> void* {
    void* p = (void*)wsp;
    wsp += (bytes + 255) & ~(size_t)255;
    return p;
  };
  float*    xl     = (float*)   carve((size_t)NNODES * 4096 * sizeof(float));
  float*    xr     = (float*)   carve((size_t)NNODES * 4096 * sizeof(float));
  float*    hA     = (float*)   carve((size_t)NNODES * 1024 * sizeof(float));
  float*    hB     = (float*)   carve((size_t)NNODES * 1024 * sizeof(float));
  float*    accum  = (float*)   carve((size_t)NNODES * 1024 * sizeof(float));
  float*    logits = (float*)   carve((size_t)ETOT * NHEADS * sizeof(float));
  float*    exv    = (float*)   carve((size_t)ETOT * NHEADS * sizeof(float));
  float*    denom  = (float*)   carve((size_t)NNODES * NHEADS * sizeof(float));
  unsigned* mkey   = (unsigned*)carve((size_t)NNODES * NHEADS * sizeof(unsigned));
  int*      srcA   = (int*)     carve((size_t)ETOT * sizeof(int));
  int*      dstA   = (int*)     carve((size_t)ETOT * sizeof(int));

  k_build_edges<<<(ETOT + 255) / 256, 256, 0, stream>>>(ei, srcA, dstA);

  const long EH = (long)ETOT * NHEADS;        // 260000 (edge,head) pairs
  const float* hin = x;
  float* hout = hA;

  for (int li = 0; li < NLAYERS; ++li) {
    const int din  = DIN[li];
    const int dout = DOUT[li];
    const int Cols = NHEADS * dout;
    const float* Wl  = (const float*)d_in[2 + 4 * li + 0];
    const float* Wr  = (const float*)d_in[2 + 4 * li + 1];
    const float* att = (const float*)d_in[2 + 4 * li + 2];
    const float* bb  = (const float*)d_in[2 + 4 * li + 3];

    // GEMMs: xl = hin@Wl, xr = hin@Wr  (grid.z selects the matrix)
    const int tiles_m = (NNODES + 15) / 16;      // 313
    const int super_n = (Cols + 31) / 32;        // 32-col supertiles
    dim3 ggrid((unsigned)super_n, (unsigned)((tiles_m + 7) / 8), 2);
    k_gemm_wmma<<<ggrid, 256, 0, stream>>>(hin, Wl, Wr, xl, xr,
                                           NNODES, din, Cols);

    const long na = (long)NNODES * dout;
    k_layer_init<<<(unsigned)((na + 255) / 256), 256, 0, stream>>>(
        accum, na, denom, mkey, NNODES * NHEADS);

    k_logits<<<(unsigned)((EH + 7) / 8), 256, 0, stream>>>(
        xl, xr, att, srcA, dstA, logits, mkey, dout);

    k_exp<<<(unsigned)((EH + 255) / 256), 256, 0, stream>>>(
        logits, mkey, denom, dstA, exv);

    k_aggr<<<(unsigned)((EH + 7) / 8), 256, 0, stream>>>(
        xl, exv, denom, srcA, dstA, accum, dout);

    float* dstbuf = (li == NLAYERS - 1) ? (float*)d_out : hout;
    k_final<<<(unsigned)((na + 255) / 256), 256, 0, stream>>>(
        accum, bb, dstbuf, na, dout, li == NLAYERS - 1 ? 1 : 0);

    hin = dstbuf;
    hout = (hout == hA) ? hB : hA;
  }
}